// MAPKSimulator_7825430414073
// MI455X (gfx1250) — compile-verified
//
#include <hip/hip_runtime.h>
#include <math.h>

// Problem constants (from reference)
#define BTRAJ    32768
#define HCTRL    10
#define NSAVE    121
#define NINT     120
#define SUBSTEPS 8
#define DT_STEP  0.0625f          // 60 / (120*8)
#define KCAT     60.0f            // 1.0 * S_TO_MIN

// Fast reciprocal: v_rcp_f32 + one Newton-Raphson refinement (~0.5 ulp).
// Denominators are K_M + concentration, strictly positive and well-conditioned.
__device__ __forceinline__ float frcp(float x) {
    float r = __builtin_amdgcn_rcpf(x);
    return r * (2.0f - x * r);
}

struct Params {
    float p0, p1, p2;      // totals: MKKK_T, MKK_T, MAPK_T
    float KM;              // params[3]
    float VMKKK, VMKK, VMAPK;  // params[4..6] * 60 * 0.05 = *3
    float p7, p8;          // E1 targets
};

// MAPK cascade vector field. 8 unique reciprocals, shared where reused.
__device__ __forceinline__ void vf(const float y[6], float u, const Params& p,
                                   float d[6]) {
    float MKKK = p.p0 - y[0];
    float MKK  = p.p1 - y[1] - y[2];
    float MAPK = p.p2 - y[3] - y[4];

    float rA = frcp(p.KM + MKKK);
    float r0 = frcp(p.KM + y[0]);
    float rB = frcp(p.KM + MKK);
    float r1 = frcp(p.KM + y[1]);
    float r2 = frcp(p.KM + y[2]);
    float rC = frcp(p.KM + MAPK);
    float r3 = frcp(p.KM + y[3]);
    float r4 = frcp(p.KM + y[4]);

    d[0] = KCAT * y[5] * MKKK * rA - p.VMKKK * y[0] * r0;

    float f1 = KCAT * y[0] * MKK  * rB;
    float f2 = KCAT * y[0] * y[1] * r1;
    float b2 = p.VMKK * y[2] * r2;
    float b1 = p.VMKK * y[1] * r1;
    d[1] = f1 - f2 + b2 - b1;
    d[2] = f2 - b2;

    float f3 = KCAT * y[2] * MAPK * rC;
    float f4 = KCAT * y[2] * y[3] * r3;
    float b4 = p.VMAPK * y[4] * r4;
    float b3 = p.VMAPK * y[3] * r3;
    d[3] = f3 - f4 + b4 - b3;
    d[4] = f4 - b4;

    // K_E1 == 1
    d[5] = fmaf(u, (p.p8 - p.p7), p.p7) - y[5];
}

__device__ __forceinline__ void rk4_step(float y[6], float u0, float uh,
                                         float u1, const Params& p) {
    float k1[6], k2[6], k3[6], k4[6], yt[6];
    vf(y, u0, p, k1);
#pragma unroll
    for (int i = 0; i < 6; ++i) yt[i] = fmaf(0.5f * DT_STEP, k1[i], y[i]);
    vf(yt, uh, p, k2);
#pragma unroll
    for (int i = 0; i < 6; ++i) yt[i] = fmaf(0.5f * DT_STEP, k2[i], y[i]);
    vf(yt, uh, p, k3);
#pragma unroll
    for (int i = 0; i < 6; ++i) yt[i] = fmaf(DT_STEP, k3[i], y[i]);
    vf(yt, u1, p, k4);
#pragma unroll
    for (int i = 0; i < 6; ++i) {
        float s = k1[i] + 2.0f * (k2[i] + k3[i]) + k4[i];
        y[i] = fmaf(DT_STEP * (1.0f / 6.0f), s, y[i]);
    }
}

// Branch-free finite filter at store time only (reference semantics:
// integration continues on raw state; only saved snapshots are cleaned).
__device__ __forceinline__ void store6(float* dst, const float y[6],
                                       float sent5) {
    const float inf = __builtin_inff();
#pragma unroll
    for (int i = 0; i < 6; ++i) {
        float v = y[i];
        float s = (i == 5) ? sent5 : 0.0f;
        dst[i] = (fabsf(v) < inf) ? v : s;
    }
}

__global__ void __launch_bounds__(64)
mapk_rk4_kernel(const float* __restrict__ init,
                const float* __restrict__ ctrl,
                const float* __restrict__ params,
                float* __restrict__ out) {
    const int b = blockIdx.x * blockDim.x + threadIdx.x;

    Params p;
    p.p0 = params[0];
    p.p1 = params[1];
    p.p2 = params[2];
    p.KM = params[3];
    p.VMKKK = params[4] * 3.0f;   // * S_TO_MIN * PHOS_SCALE
    p.VMKK  = params[5] * 3.0f;
    p.VMAPK = params[6] * 3.0f;
    p.p7 = params[7];
    p.p8 = params[8];
    const float sent5 = p.p7;

    float y[6];
#pragma unroll
    for (int i = 0; i < 6; ++i) y[i] = init[b * 6 + i];

    float* ob = out + (size_t)b * (NSAVE * 6);
    store6(ob, y, sent5);   // snapshot 0 = (filtered) initial state

    const float* cb = ctrl + b * HCTRL;
    int n = 0;

    // Control segment c covers 96 substeps = 12 save intervals.
    // Within it u0 == uh == u[c]; u1 == u[c] except the segment's final
    // substep, where u1 = u[min(c+1, 9)].
#pragma unroll 1
    for (int c = 0; c < HCTRL; ++c) {
        float uc = fminf(fmaxf(cb[c], 0.0f), 1.0f);
        int cn = (c < HCTRL - 1) ? (c + 1) : (HCTRL - 1);
        float un = fminf(fmaxf(cb[cn], 0.0f), 1.0f);

#pragma unroll 1
        for (int j = 0; j < 12; ++j) {
#pragma unroll 1
            for (int s = 0; s < SUBSTEPS; ++s) {
                float u1 = (j == 11 && s == SUBSTEPS - 1) ? un : uc;
                rk4_step(y, uc, uc, u1, p);
            }
            ++n;
            store6(ob + n * 6, y, sent5);
        }
    }
}

extern "C" void kernel_launch(void* const* d_in, const int* in_sizes, int n_in,
                              void* d_out, int out_size, void* d_ws,
                              size_t ws_size, hipStream_t stream) {
    const float* initial_state = (const float*)d_in[0];  // (B, 6)
    const float* control_seq   = (const float*)d_in[1];  // (B, H, 1)
    const float* params        = (const float*)d_in[2];  // (9,)
    float* out = (float*)d_out;                          // (B, 121, 6)

    const int block = 64;                 // 2 waves/WG -> 512 WGs spread wide
    const int grid  = BTRAJ / block;
    mapk_rk4_kernel<<<grid, block, 0, stream>>>(initial_state, control_seq,
                                                params, out);
}